// Mamba1DConv_29300266893416
// MI455X (gfx1250) — compile-verified
//
#include <hip/hip_runtime.h>
#include <cstdint>
#include <cstddef>

// Problem constants (from reference): B=2, L=4096, D=4096, K=4 (causal depthwise).
#define B_  2
#define L_  4096
#define D_  4096
#define K_  4
#define TL  8               // L rows produced per block
#define TD  1024            // D floats per block (256 threads * float4)
#define NROWS (TL + K_ - 1) // 11 staged rows (includes 3-row causal halo)
#define THREADS 256

typedef __attribute__((ext_vector_type(4))) float f4;
typedef __attribute__((ext_vector_type(4))) int   v4i;

// CDNA5 async global->LDS staging path (ASYNCcnt-tracked), guarded so the
// kernel still compiles (with plain b128 loads) if the builtin is absent.
#if defined(__has_builtin)
#if __has_builtin(__builtin_amdgcn_global_load_async_to_lds_b128) && \
    __has_builtin(__builtin_amdgcn_s_wait_asynccnt)
#define USE_ASYNC_LDS 1
#endif
#endif
#ifndef USE_ASYNC_LDS
#define USE_ASYNC_LDS 0
#endif

__device__ __forceinline__ void stage_row(const float* __restrict__ src,
                                          float* __restrict__ dst_lds)
{
#if USE_ASYNC_LDS
    __builtin_amdgcn_global_load_async_to_lds_b128(
        (__attribute__((address_space(1))) v4i*)src,
        (__attribute__((address_space(3))) v4i*)dst_lds,
        /*offset=*/0, /*cpol=*/0);
#else
    *(f4*)dst_lds = *(const f4*)src;
#endif
}

__global__ __launch_bounds__(THREADS)
void mamba_dwconv_kernel(const float* __restrict__ x,
                         const float* __restrict__ w,
                         const float* __restrict__ bias,
                         float* __restrict__ y)
{
    __shared__ float lds[NROWS * TD];   // 11 rows * 4KB = 44KB

    const int tid   = threadIdx.x;                 // 0..255
    const int blk   = blockIdx.x;                  // 0..4095
    const int dTile = blk & 3;                     // 4 D-tiles
    const int lTile = (blk >> 2) & (L_ / TL - 1);  // 512 L-tiles
    const int batch = blk >> 11;                   // 2 batches

    const int d  = dTile * TD + tid * 4;           // this thread's 4 channels
    const int l0 = lTile * TL;

    const size_t bstride = (size_t)L_ * D_;
    const float* xb = x + (size_t)batch * bstride;
    float*       yb = y + (size_t)batch * bstride;

    const float* src0 = xb + (size_t)(l0 - (K_ - 1)) * D_ + d; // row l0-3 (maybe OOB)
    float*       lds0 = &lds[tid * 4];

    // ---- Stage rows l0-3 .. l0+TL-1 of this D-tile into LDS (b128 per thread/row).
    if (l0 >= K_ - 1) {
        // Steady state (4088 of 4096 blocks): no guards, 11 back-to-back async loads.
#pragma unroll
        for (int r = 0; r < NROWS; ++r)
            stage_row(src0 + (size_t)r * D_, lds0 + r * TD);
    } else {
        // lTile == 0: first K-1 rows are causal zero padding.
        f4 z = {0.f, 0.f, 0.f, 0.f};
#pragma unroll
        for (int r = 0; r < K_ - 1; ++r)
            *(f4*)(lds0 + r * TD) = z;
#pragma unroll
        for (int r = K_ - 1; r < NROWS; ++r)
            stage_row(src0 + (size_t)r * D_, lds0 + r * TD);
    }

    // Warm L2 for the next L-tile (speculative prefetch; harmless at the edge).
    if (l0 + TL < L_) {
        __builtin_prefetch(xb + (size_t)(l0 + TL) * D_ + d, 0, 0);
    }

    // ---- Per-channel taps and bias (tiny, L2-resident).
    const f4 wch0 = *(const f4*)(w + (size_t)(d + 0) * K_);  // taps k0..k3, channel d+0
    const f4 wch1 = *(const f4*)(w + (size_t)(d + 1) * K_);
    const f4 wch2 = *(const f4*)(w + (size_t)(d + 2) * K_);
    const f4 wch3 = *(const f4*)(w + (size_t)(d + 3) * K_);
    const f4 bv   = *(const f4*)(bias + d);

#if USE_ASYNC_LDS
    __builtin_amdgcn_s_wait_asynccnt(0);           // async LDS writes complete
#endif
    __syncthreads();                               // publish staged tile block-wide

    // ---- Sliding 4-row window from LDS; 8 outputs per thread, 16 FMA each.
    f4 r0 = *(const f4*)(lds0 + 0 * TD);           // l0-3
    f4 r1 = *(const f4*)(lds0 + 1 * TD);           // l0-2
    f4 r2 = *(const f4*)(lds0 + 2 * TD);           // l0-1
#pragma unroll
    for (int t = 0; t < TL; ++t) {
        const f4 r3 = *(const f4*)(lds0 + (t + 3) * TD);         // l0+t
        f4 o;
        o.x = fmaf(wch0.x, r0.x, fmaf(wch0.y, r1.x, fmaf(wch0.z, r2.x, fmaf(wch0.w, r3.x, bv.x))));
        o.y = fmaf(wch1.x, r0.y, fmaf(wch1.y, r1.y, fmaf(wch1.z, r2.y, fmaf(wch1.w, r3.y, bv.y))));
        o.z = fmaf(wch2.x, r0.z, fmaf(wch2.y, r1.z, fmaf(wch2.z, r2.z, fmaf(wch2.w, r3.z, bv.z))));
        o.w = fmaf(wch3.x, r0.w, fmaf(wch3.y, r1.w, fmaf(wch3.z, r2.w, fmaf(wch3.w, r3.w, bv.w))));
        // Output is never re-read: non-temporal b128 store, keep L2 for halo reads.
        __builtin_nontemporal_store(o, (f4*)(yb + (size_t)(l0 + t) * D_ + d));
        r0 = r1; r1 = r2; r2 = r3;
    }
}

extern "C" void kernel_launch(void* const* d_in, const int* in_sizes, int n_in,
                              void* d_out, int out_size, void* d_ws, size_t ws_size,
                              hipStream_t stream) {
    const float* x    = (const float*)d_in[0];   // [B, L, D]
    const float* w    = (const float*)d_in[1];   // [D, 1, K] contiguous -> w[d*K + k]
    const float* bias = (const float*)d_in[2];   // [D]
    float*       y    = (float*)d_out;           // [B, L, D]

    const int grid = B_ * (L_ / TL) * (D_ / TD); // 2 * 512 * 4 = 4096 blocks
    mamba_dwconv_kernel<<<grid, THREADS, 0, stream>>>(x, w, bias, y);
}